// FCCapsuleLayer_69363721831091
// MI455X (gfx1250) — compile-verified
//
#include <hip/hip_runtime.h>

typedef __attribute__((ext_vector_type(2))) float v2f;
typedef __attribute__((ext_vector_type(8))) float v8f;

#define ICAPS   32
#define IATOMS  16
#define NCAPS   10
#define CDIM    16
#define NCD     160          // NCAPS*CDIM
#define NROUT   3
#define EPSV    1e-7f
#define XSTR    17           // padded stride for sX rows  (gcd(17,64)=1 -> conflict free)
#define WSTR    164          // padded stride for sW/sV rows (breaks half-wave aliasing)

__global__ __launch_bounds__(160)
void caps_routing_fused(const float* __restrict__ x,    // [8192, 32, 16]
                        const float* __restrict__ Wm,   // [16, 160]
                        const float* __restrict__ bias, // [160]
                        float* __restrict__ out)        // [8192, 160]
{
    __shared__ float sX[ICAPS * XSTR];        // 544  f
    __shared__ float sW[IATOMS * WSTR];       // 2624 f
    __shared__ float sV[ICAPS * WSTR];        // 5248 f  (votes, padded rows)
    __shared__ float sRoute[ICAPS * NCAPS];   // 320  f
    __shared__ float sLogit[ICAPS * NCAPS];   // 320  f
    __shared__ float sAct[NCD];               // 160  f

    const int t    = threadIdx.x;
    const int lane = t & 31;
    const int wave = t >> 5;
    const int half = lane >> 4;      // 0 | 1 : which K/M half of the wave
    const int lcol = lane & 15;      // N (or M for A-frag) index within tile
    const long pos = blockIdx.x;     // (b,h,w) flattened

    // ---- stage inputs into LDS (padded layouts) ----
    const float* xg = x + pos * (ICAPS * IATOMS);
    for (int i = t; i < ICAPS * IATOMS; i += 160)
        sX[(i >> 4) * XSTR + (i & 15)] = xg[i];
    for (int i = t; i < IATOMS * NCD; i += 160)
        sW[(i / NCD) * WSTR + (i % NCD)] = Wm[i];
    for (int i = t; i < ICAPS * NCAPS; i += 160)
        sLogit[i] = 0.0f;
    const float bval = bias[t];
    __syncthreads();

    // ---- votes = X(32x16) * W(16x160) via V_WMMA_F32_16X16X4_F32 ----
    // 20 tiles (2 M-tiles x 10 N-tiles), 5 waves -> 4 tiles per wave.
    for (int i = 0; i < 4; ++i) {
        const int tile  = wave * 4 + i;       // uniform per wave
        const int mtile = tile / NCAPS;
        const int ntile = tile % NCAPS;
        v8f acc = {};
#pragma unroll
        for (int k0 = 0; k0 < IATOMS; k0 += 4) {
            // A frag: lane l -> M = lcol (+16*mtile), K = k0 + 2*half + v
            const float* ap = &sX[(mtile * 16 + lcol) * XSTR + k0 + 2 * half];
            v2f a; a.x = ap[0]; a.y = ap[1];
            // B frag: lane l -> K = k0 + 2*half + v, N = ntile*16 + lcol
            const float* bp = &sW[(k0 + 2 * half) * WSTR + ntile * 16 + lcol];
            v2f b; b.x = bp[0]; b.y = bp[WSTR];
            acc = __builtin_amdgcn_wmma_f32_16x16x4_f32(
                      false, a, false, b, (short)0, acc, false, false);
        }
        // D frag: lane l, vgpr v -> M = mtile*16 + 8*half + v, N = ntile*16 + lcol
        float* vd = &sV[(mtile * 16 + 8 * half) * WSTR + ntile * 16 + lcol];
#pragma unroll
        for (int v = 0; v < 8; ++v) vd[v * WSTR] = acc[v];
    }
    __syncthreads();

    // ---- dynamic routing (3 iterations), all in LDS/registers ----
    const int nc = t >> 4;           // output capsule of this thread
    float act = 0.0f;
    for (int r = 0; r < NROUT; ++r) {
        // softmax over NC per input capsule: wave 0, one ic per lane (uniform branch)
        if (t < ICAPS) {
            float lg[NCAPS], mx = -1e30f;
#pragma unroll
            for (int n = 0; n < NCAPS; ++n) { lg[n] = sLogit[t * NCAPS + n]; mx = fmaxf(mx, lg[n]); }
            float s = 0.0f;
#pragma unroll
            for (int n = 0; n < NCAPS; ++n) { lg[n] = __expf(lg[n] - mx); s += lg[n]; }
            const float inv = 1.0f / s;
#pragma unroll
            for (int n = 0; n < NCAPS; ++n) sRoute[t * NCAPS + n] = lg[n] * inv;
        }
        __syncthreads();

        // preactivate[nc,cd] = sum_ic route[ic,nc] * votes[ic,nc,cd] + bias
        float p = bval;
#pragma unroll 4
        for (int ic = 0; ic < ICAPS; ++ic)
            p += sRoute[ic * NCAPS + nc] * sV[ic * WSTR + t];

        // squash over CD=16 (width-16 xor reduction inside the wave32)
        float sq = p * p;
        sq += __shfl_xor(sq, 1, 16);
        sq += __shfl_xor(sq, 2, 16);
        sq += __shfl_xor(sq, 4, 16);
        sq += __shfl_xor(sq, 8, 16);
        act = (sq / (1.0f + sq)) * p * rsqrtf(sq + EPSV);
        sAct[t] = act;
        __syncthreads();

        if (r < NROUT - 1) {
            // logits[ic,nc] += dot(votes[ic,nc,:], act[nc,:]) ; 320 pairs, 2/thread
            for (int pr = t; pr < ICAPS * NCAPS; pr += 160) {
                const int ic = pr / NCAPS, n = pr % NCAPS;
                const float* vr = &sV[ic * WSTR + n * CDIM];
                const float* ar = &sAct[n * CDIM];
                float s = 0.0f;
#pragma unroll
                for (int c = 0; c < CDIM; ++c) s += vr[c] * ar[c];
                sLogit[pr] += s;
            }
            __syncthreads();
        }
    }

    out[pos * NCD + t] = act;
}

extern "C" void kernel_launch(void* const* d_in, const int* in_sizes, int n_in,
                              void* d_out, int out_size, void* d_ws, size_t ws_size,
                              hipStream_t stream) {
    const float* x  = (const float*)d_in[0];   // [32,16,16,32,16]
    const float* W  = (const float*)d_in[1];   // [16,160]
    const float* b  = (const float*)d_in[2];   // [1,1,10,16]
    float* out      = (float*)d_out;           // [32,16,16,10,16]
    const int positions = 32 * 16 * 16;        // 8192
    caps_routing_fused<<<dim3(positions), dim3(160), 0, stream>>>(x, W, b, out);
}